// Communication_89464168775793
// MI455X (gfx1250) — compile-verified
//
#include <hip/hip_runtime.h>

typedef __attribute__((ext_vector_type(2))) float v2f;
typedef __attribute__((ext_vector_type(4))) float v4f;
typedef __attribute__((ext_vector_type(8))) float v8f;

#define C_S 384
#define C_H 128
#define C_Z 128
#define NROW 512

// ---------------------------------------------------------------------------
// Kernel 1: fused [X @ Wᵀ + bias] -> LayerNorm -> [a @ WoutHalfᵀ] * (1/D) (+bout)
// One block = one 16-row strip of one branch (sel 0: s1->P', sel 1: s2->Q').
// 8 waves per block; wave w computes the 16x16 tile for column block w using
// v_wmma_f32_16x16x4_f32 (fp32 WMMA, matches reference precision).
// ---------------------------------------------------------------------------
__global__ __launch_bounds__(256) void proj_ln_half_kernel(
    const float* __restrict__ s1, const float* __restrict__ s2,
    const float* __restrict__ W1, const float* __restrict__ b1,
    const float* __restrict__ W2, const float* __restrict__ b2,
    const float* __restrict__ gamma, const float* __restrict__ beta,
    const float* __restrict__ Wout, const float* __restrict__ bout,
    float* __restrict__ Pp, float* __restrict__ Qp, float invD)
{
    const int sel   = blockIdx.x >> 5;     // 0: s1 branch, 1: s2 branch
    const int strip = blockIdx.x & 31;     // which 16-row strip of 512
    const int tid   = threadIdx.x;
    const int wave  = tid >> 5;            // 0..7 -> 16-column block
    const int lane  = tid & 31;
    const int half  = lane >> 4;           // K-pair select (ISA A/B layout)
    const int lid   = lane & 15;           // row (A) / col (B) within tile

    const float* __restrict__ X    = sel ? s2 : s1;
    const float* __restrict__ W    = sel ? W2 : W1;
    const float* __restrict__ bias = sel ? b2 : b1;

    const int row0 = strip * 16;

    __shared__ float ln[16][C_H + 4];
    __shared__ float s_mu[16], s_rs[16];

    // ---- GEMM1: Y[16 x 128] = X_strip @ Wᵀ ; wave owns cols [wave*16, +16)
    v8f acc = {};
    const float* __restrict__ Arow = X + (row0 + lid) * C_S;      // A: M=lid
    const float* __restrict__ Brow = W + (wave * 16 + lid) * C_S; // B[k][n]=W[n][k]
    for (int kk = 0; kk < C_S; kk += 4) {
        const int ko = kk + 2 * half;                 // lanes 0-15: K0,K1; 16-31: K2,K3
        v2f af = { Arow[ko], Arow[ko + 1] };
        v2f bf = { Brow[ko], Brow[ko + 1] };
        acc = __builtin_amdgcn_wmma_f32_16x16x4_f32(false, af, false, bf,
                                                    (short)0, acc, false, false);
    }

    // scatter C tile + bias into LDS (C layout: VGPR r -> rows r / r+8)
    {
        const int col = wave * 16 + lid;
        const float bv = bias[col];
#pragma unroll
        for (int r = 0; r < 8; ++r)
            ln[r + 8 * half][col] = acc[r] + bv;
    }
    __syncthreads();

    // ---- LayerNorm stats: one thread per row (16 rows x 128 cols, tiny)
    if (tid < 16) {
        float mu = 0.f;
        for (int c = 0; c < C_H; ++c) mu += ln[tid][c];
        mu *= (1.0f / C_H);
        float var = 0.f;
        for (int c = 0; c < C_H; ++c) { float d = ln[tid][c] - mu; var += d * d; }
        var *= (1.0f / C_H);
        s_mu[tid] = mu;
        s_rs[tid] = rsqrtf(var + 1e-5f);
    }
    __syncthreads();

    // normalize in place: 256 threads x 8 elements = 2048
#pragma unroll
    for (int j = 0; j < 8; ++j) {
        const int e = tid * 8 + j;
        const int r = e >> 7, c = e & 127;
        ln[r][c] = (ln[r][c] - s_mu[r]) * s_rs[r] * gamma[c] + beta[c];
    }
    __syncthreads();

    // ---- GEMM2: T[16 x 128] = a_strip @ WoutHalfᵀ  (K = 128, from LDS)
    v8f acc2 = {};
    const int zc = wave * 16 + lid;                               // output z channel
    const float* __restrict__ Wo = Wout + zc * (2 * C_H) + sel * C_H;
    for (int kk = 0; kk < C_H; kk += 4) {
        const int ko = kk + 2 * half;
        v2f af = { ln[lid][ko], ln[lid][ko + 1] };
        v2f bf = { Wo[ko], Wo[ko + 1] };
        acc2 = __builtin_amdgcn_wmma_f32_16x16x4_f32(false, af, false, bf,
                                                     (short)0, acc2, false, false);
    }

    // scale by 1/D; fold bout into the Q' branch
    {
        float* __restrict__ Out = sel ? Qp : Pp;
        const float extra = sel ? bout[zc] : 0.0f;
#pragma unroll
        for (int r = 0; r < 8; ++r) {
            const int grow = row0 + r + 8 * half;
            Out[grow * C_Z + zc] = acc2[r] * invD + extra;
        }
    }
}

// ---------------------------------------------------------------------------
// Kernel 2: out[n,m,z] = P'[n,z] + Q'[m,z]  — pure store-bandwidth bound
// (128 MiB streamed with non-temporal float4 stores; P'/Q' are L2-resident).
// Block = one n and 32 consecutive m; thread: z = (tid&31)*4, m_off = tid>>5.
// ---------------------------------------------------------------------------
__global__ __launch_bounds__(256) void bcast_kernel(
    const float* __restrict__ Pp, const float* __restrict__ Qp,
    float* __restrict__ out)
{
    const int n    = blockIdx.x >> 4;
    const int mb   = blockIdx.x & 15;
    const int tid  = threadIdx.x;
    const int z    = (tid & 31) * 4;
    const int moff = tid >> 5;

    const v4f p = *(const v4f*)(Pp + n * C_Z + z);
    float* __restrict__ base = out + (size_t)n * NROW * C_Z;
#pragma unroll
    for (int i = 0; i < 4; ++i) {
        const int m = mb * 32 + moff + i * 8;
        const v4f q = *(const v4f*)(Qp + m * C_Z + z);
        v4f v = p + q;
        __builtin_nontemporal_store(v, (v4f*)(base + (size_t)m * C_Z + z));
    }
}

extern "C" void kernel_launch(void* const* d_in, const int* in_sizes, int n_in,
                              void* d_out, int out_size, void* d_ws, size_t ws_size,
                              hipStream_t stream) {
    const float* s1    = (const float*)d_in[0];
    const float* s2    = (const float*)d_in[1];
    const float* W1    = (const float*)d_in[2];
    const float* b1    = (const float*)d_in[3];
    const float* W2    = (const float*)d_in[4];
    const float* b2    = (const float*)d_in[5];
    const float* gamma = (const float*)d_in[6];
    const float* beta  = (const float*)d_in[7];
    const float* Wout  = (const float*)d_in[8];
    const float* bout  = (const float*)d_in[9];
    float* out = (float*)d_out;

    float* Pp = (float*)d_ws;                 // 512*128 floats
    float* Qp = Pp + NROW * C_Z;              // 512*128 floats (bout folded in)

    const float invD = 1.0f / (0.001f + 512.0f * 512.0f);

    proj_ln_half_kernel<<<64, 256, 0, stream>>>(s1, s2, W1, b1, W2, b2,
                                                gamma, beta, Wout, bout,
                                                Pp, Qp, invD);
    bcast_kernel<<<8192, 256, 0, stream>>>(Pp, Qp, out);
}